// GCN_64493228916895
// MI455X (gfx1250) — compile-verified
//
#include <hip/hip_runtime.h>

// GCN (3x GCNConv, D=128) for MI455X / gfx1250.
// Phases: [once] deg -> dinv -> edge norms; [per layer] f32-WMMA GEMM,
// self-loop init, edge scatter with global_atomic_add_f32, fused bias+ReLU.

#define D_FEAT 128

typedef __attribute__((ext_vector_type(2))) float v2f;
typedef __attribute__((ext_vector_type(8))) float v8f;

// ---------------------------------------------------------------- fill
__global__ __launch_bounds__(256) void k_fill(float* __restrict__ p, float v, int n) {
    int i = blockIdx.x * blockDim.x + threadIdx.x;
    if (i < n) p[i] = v;
}

// ------------------------------------------------------------- degree
__global__ __launch_bounds__(256) void k_deg(const long long* __restrict__ dst,
                                             const float* __restrict__ w,
                                             float* __restrict__ deg, int ne) {
    int e = blockIdx.x * blockDim.x + threadIdx.x;
    if (e < ne) unsafeAtomicAdd(&deg[dst[e]], w[e]);
}

__global__ __launch_bounds__(256) void k_rsqrt_ip(float* __restrict__ d, int n) {
    int i = blockIdx.x * blockDim.x + threadIdx.x;
    if (i < n) { float v = d[i]; d[i] = v > 0.f ? rsqrtf(v) : 0.f; }
}

// ----------------------------------------------------------- edge norm
__global__ __launch_bounds__(256) void k_norm(const long long* __restrict__ src,
                                              const long long* __restrict__ dst,
                                              const float* __restrict__ w,
                                              const float* __restrict__ dinv,
                                              float* __restrict__ norm, int ne) {
    int e = blockIdx.x * blockDim.x + threadIdx.x;
    if (e < ne) norm[e] = dinv[src[e]] * w[e] * dinv[dst[e]];
}

// ------------------------------------------------- GEMM: H = X @ W  (f32 WMMA)
// One wave computes one 16x16 tile of H; 8 waves/block cover the 8 N-tiles of
// an M-tile. K=128 swept with 32x V_WMMA_F32_16X16X4_F32 (fully unrolled).
__global__ __launch_bounds__(256) void k_gemm(const float* __restrict__ X,
                                              const float* __restrict__ W,
                                              float* __restrict__ H, int M) {
    const int lane = threadIdx.x & 31;
    const int half = lane >> 4;      // 0: K pair {0,1}; 1: K pair {2,3}
    const int lr   = lane & 15;
    const int n0   = (threadIdx.x >> 5) * 16;   // wave id -> N tile
    const int m0   = blockIdx.x * 16;
    if (m0 + 16 > M) return;  // uniform branch; M % 16 == 0 in practice

    const float* xr = X + (size_t)(m0 + lr) * D_FEAT;
    v8f c = {};
#pragma unroll
    for (int k = 0; k < D_FEAT; k += 4) {
        const int kk = k + 2 * half;
        v2f a, b;
        // A 16x4 f32: lanes 0-15 hold K={k,k+1}, lanes 16-31 hold K={k+2,k+3}
        a.x = xr[kk];
        a.y = xr[kk + 1];
        // B 4x16 f32 (row-major W[128][128]): same half-wave K split, N across lanes
        b.x = W[(size_t)kk       * D_FEAT + n0 + lr];
        b.y = W[(size_t)(kk + 1) * D_FEAT + n0 + lr];
        c = __builtin_amdgcn_wmma_f32_16x16x4_f32(false, a, false, b,
                                                  (short)0, c, false, false);
    }
#pragma unroll
    for (int r = 0; r < 8; ++r)
        H[(size_t)(m0 + r + 8 * half) * D_FEAT + n0 + lr] = c[r];
}

// --------------------------------------- agg init with self-loop contribution
// agg[n,f] = dinv[n]^2 * h[n,f]   (self-loop weight 1.0, avoids N*128 atomics)
__global__ __launch_bounds__(256) void k_selfinit(const float* __restrict__ dinv,
                                                  const float* __restrict__ h,
                                                  float* __restrict__ agg, int nf) {
    int i = blockIdx.x * blockDim.x + threadIdx.x;
    if (i < nf) {
        float d = dinv[i >> 7];
        agg[i] = d * d * h[i];
    }
}

// ------------------------------------------------ edge scatter (atomic f32)
// One wave per edge: lane i gathers h[src][4i..4i+3] (float4, coalesced 512B)
// and atomically accumulates norm*h into agg[dst]. Gathers hit L2 (51MB << 192MB).
__global__ __launch_bounds__(256) void k_scatter(const long long* __restrict__ src,
                                                 const long long* __restrict__ dst,
                                                 const float* __restrict__ norm,
                                                 const float* __restrict__ H,
                                                 float* __restrict__ AGG, int ne) {
    int e = blockIdx.x * 8 + (threadIdx.x >> 5);
    if (e >= ne) return;
    const int lane = threadIdx.x & 31;
    const long long s = src[e];
    const long long d = dst[e];
    const float nv = norm[e];
    const float4 hv = ((const float4*)(H + (size_t)s * D_FEAT))[lane];
    float* o = AGG + (size_t)d * D_FEAT + lane * 4;
    unsafeAtomicAdd(o + 0, nv * hv.x);
    unsafeAtomicAdd(o + 1, nv * hv.y);
    unsafeAtomicAdd(o + 2, nv * hv.z);
    unsafeAtomicAdd(o + 3, nv * hv.w);
}

// ------------------------------------------------------- bias + ReLU epilogue
__global__ __launch_bounds__(256) void k_bias_relu(const float* __restrict__ in,
                                                   const float* __restrict__ b,
                                                   float* __restrict__ out, int nf) {
    int i = blockIdx.x * blockDim.x + threadIdx.x;
    if (i < nf) out[i] = fmaxf(in[i] + b[i & (D_FEAT - 1)], 0.f);
}

// ----------------------------------------------------------------- launch
extern "C" void kernel_launch(void* const* d_in, const int* in_sizes, int n_in,
                              void* d_out, int out_size, void* d_ws, size_t ws_size,
                              hipStream_t stream) {
    const float*     x   = (const float*)d_in[0];
    const long long* ei  = (const long long*)d_in[1];
    const float*     wgt = (const float*)d_in[2];
    const float*     W0  = (const float*)d_in[3];
    const float*     b0  = (const float*)d_in[4];
    const float*     W1  = (const float*)d_in[5];
    const float*     b1  = (const float*)d_in[6];
    const float*     W2  = (const float*)d_in[7];
    const float*     b2  = (const float*)d_in[8];
    float* out = (float*)d_out;

    const int N  = in_sizes[0] / D_FEAT;   // 100000
    const int E  = in_sizes[2];            // 1600000
    const int NF = N * D_FEAT;             // 12.8M

    const long long* src = ei;
    const long long* dst = ei + E;

    // workspace layout (floats): dinv[N] | norm[E] | buf0[N*128] | buf1[N*128]
    float* ws   = (float*)d_ws;
    float* dinv = ws;
    float* norm = ws + N;
    float* buf0 = norm + E;                 // h  (GEMM output)
    float* buf1 = buf0 + (size_t)NF;        // agg / activation (in-place)

    const int BLK = 256;
    const int gN  = (N + BLK - 1) / BLK;
    const int gE  = (E + BLK - 1) / BLK;
    const int gNF = (NF + BLK - 1) / BLK;
    const int gEw = (E + 7) / 8;            // wave-per-edge scatter
    const int gM  = (N + 15) / 16;          // GEMM M tiles

    // --- graph structure (once): deg -> dinv -> edge norms ---
    k_fill<<<gN, BLK, 0, stream>>>(dinv, 1.0f, N);                // self-loop wt
    k_deg<<<gE, BLK, 0, stream>>>(dst, wgt, dinv, E);
    k_rsqrt_ip<<<gN, BLK, 0, stream>>>(dinv, N);
    k_norm<<<gE, BLK, 0, stream>>>(src, dst, wgt, dinv, norm, E);

    // --- layer 0: x -> buf1 ---
    k_gemm<<<gM, BLK, 0, stream>>>(x, W0, buf0, N);
    k_selfinit<<<gNF, BLK, 0, stream>>>(dinv, buf0, buf1, NF);
    k_scatter<<<gEw, BLK, 0, stream>>>(src, dst, norm, buf0, buf1, E);
    k_bias_relu<<<gNF, BLK, 0, stream>>>(buf1, b0, buf1, NF);

    // --- layer 1: buf1 -> buf1 (h via buf0) ---
    k_gemm<<<gM, BLK, 0, stream>>>(buf1, W1, buf0, N);
    k_selfinit<<<gNF, BLK, 0, stream>>>(dinv, buf0, buf1, NF);
    k_scatter<<<gEw, BLK, 0, stream>>>(src, dst, norm, buf0, buf1, E);
    k_bias_relu<<<gNF, BLK, 0, stream>>>(buf1, b1, buf1, NF);

    // --- layer 2: buf1 -> d_out ---
    k_gemm<<<gM, BLK, 0, stream>>>(buf1, W2, buf0, N);
    k_selfinit<<<gNF, BLK, 0, stream>>>(dinv, buf0, buf1, NF);
    k_scatter<<<gEw, BLK, 0, stream>>>(src, dst, norm, buf0, buf1, E);
    k_bias_relu<<<gNF, BLK, 0, stream>>>(buf1, b2, out, NF);
}